// BinarizeLayer_14680198217839
// MI455X (gfx1250) — compile-verified
//
#include <hip/hip_runtime.h>
#include <hip/hip_bf16.h>
#include <stdint.h>

// out[b,f] = (medians[f] > 0) && (inputs[b,f] >= medians[f]), output as f32 0/1.
// Memory-bound: ~268 MB traffic -> ~11.5 us floor at 23.3 TB/s. Strategy:
//  - b128 non-temporal streaming loads/stores (float4/lane, 512 B per wave32 op)
//  - medians (16 KB) staged to LDS per block via gfx1250 async copy
//    (global_load_async_to_lds_b128 + s_wait_asynccnt), read back via ds_load_b128
//  - launch stride is a multiple of ncols4 => per-thread column is loop-invariant;
//    fast path hoists the median read out of the loop and 4x-unrolls the stream.

typedef float v4f __attribute__((ext_vector_type(4)));

#define SMED_VEC4 1024  // 1024 x float4 = 16 KB LDS: full 4096-entry medians row

__global__ __launch_bounds__(256) void binarize_vec_kernel(
    const float* __restrict__ in, const float* __restrict__ med,
    float* __restrict__ out, int total4, int ncols4, int stride_mod)
{
    __shared__ v4f smed4[SMED_VEC4];
    const int t = threadIdx.x;
    const bool lds_ok = ncols4 <= SMED_VEC4;

    if (lds_ok) {
        // Async-stage medians into LDS: one 16-byte global->LDS async transfer
        // per lane per iteration (ASYNCcnt-tracked, no VGPR data movement).
        for (int k = t; k < ncols4; k += (int)blockDim.x) {
            uint32_t lds_off = (uint32_t)(uintptr_t)(&smed4[k]);
            uint64_t gaddr   = (uint64_t)(uintptr_t)(med + (size_t)k * 4);
            asm volatile("global_load_async_to_lds_b128 %0, %1, off"
                         :: "v"(lds_off), "v"(gaddr) : "memory");
        }
        asm volatile("s_wait_asynccnt 0" ::: "memory");
    }
    __syncthreads();

    const int gid    = (int)(blockIdx.x * blockDim.x + t);
    const int stride = (int)(gridDim.x * blockDim.x);
    int c4           = gid % ncols4;  // one-time integer division

    if (stride_mod == 0 && lds_ok) {
        // Fast path: column (and median) is loop-invariant for this thread.
        const v4f m = smed4[c4];      // ds_load_b128, once
        int i = gid;
        // 4x unrolled stream: 4 independent NT loads in flight per round.
        for (; i + 3 * stride < total4; i += 4 * stride) {
            const v4f x0 = __builtin_nontemporal_load(((const v4f*)in) + i);
            const v4f x1 = __builtin_nontemporal_load(((const v4f*)in) + i + stride);
            const v4f x2 = __builtin_nontemporal_load(((const v4f*)in) + i + 2 * stride);
            const v4f x3 = __builtin_nontemporal_load(((const v4f*)in) + i + 3 * stride);
            v4f r0, r1, r2, r3;
            r0.x = (m.x > 0.0f && x0.x >= m.x) ? 1.0f : 0.0f;
            r0.y = (m.y > 0.0f && x0.y >= m.y) ? 1.0f : 0.0f;
            r0.z = (m.z > 0.0f && x0.z >= m.z) ? 1.0f : 0.0f;
            r0.w = (m.w > 0.0f && x0.w >= m.w) ? 1.0f : 0.0f;
            r1.x = (m.x > 0.0f && x1.x >= m.x) ? 1.0f : 0.0f;
            r1.y = (m.y > 0.0f && x1.y >= m.y) ? 1.0f : 0.0f;
            r1.z = (m.z > 0.0f && x1.z >= m.z) ? 1.0f : 0.0f;
            r1.w = (m.w > 0.0f && x1.w >= m.w) ? 1.0f : 0.0f;
            r2.x = (m.x > 0.0f && x2.x >= m.x) ? 1.0f : 0.0f;
            r2.y = (m.y > 0.0f && x2.y >= m.y) ? 1.0f : 0.0f;
            r2.z = (m.z > 0.0f && x2.z >= m.z) ? 1.0f : 0.0f;
            r2.w = (m.w > 0.0f && x2.w >= m.w) ? 1.0f : 0.0f;
            r3.x = (m.x > 0.0f && x3.x >= m.x) ? 1.0f : 0.0f;
            r3.y = (m.y > 0.0f && x3.y >= m.y) ? 1.0f : 0.0f;
            r3.z = (m.z > 0.0f && x3.z >= m.z) ? 1.0f : 0.0f;
            r3.w = (m.w > 0.0f && x3.w >= m.w) ? 1.0f : 0.0f;
            __builtin_nontemporal_store(r0, ((v4f*)out) + i);
            __builtin_nontemporal_store(r1, ((v4f*)out) + i + stride);
            __builtin_nontemporal_store(r2, ((v4f*)out) + i + 2 * stride);
            __builtin_nontemporal_store(r3, ((v4f*)out) + i + 3 * stride);
        }
        for (; i < total4; i += stride) {
            const v4f x = __builtin_nontemporal_load(((const v4f*)in) + i);
            v4f r;
            r.x = (m.x > 0.0f && x.x >= m.x) ? 1.0f : 0.0f;
            r.y = (m.y > 0.0f && x.y >= m.y) ? 1.0f : 0.0f;
            r.z = (m.z > 0.0f && x.z >= m.z) ? 1.0f : 0.0f;
            r.w = (m.w > 0.0f && x.w >= m.w) ? 1.0f : 0.0f;
            __builtin_nontemporal_store(r, ((v4f*)out) + i);
        }
    } else {
        // Generic path: column varies per iteration; incremental mod update.
        for (int i = gid; i < total4; i += stride) {
            const v4f x = __builtin_nontemporal_load(((const v4f*)in) + i);
            const v4f m = lds_ok ? smed4[c4]                 // ds_load_b128
                                 : *(((const v4f*)med) + c4); // L2-resident
            v4f r;
            r.x = (m.x > 0.0f && x.x >= m.x) ? 1.0f : 0.0f;
            r.y = (m.y > 0.0f && x.y >= m.y) ? 1.0f : 0.0f;
            r.z = (m.z > 0.0f && x.z >= m.z) ? 1.0f : 0.0f;
            r.w = (m.w > 0.0f && x.w >= m.w) ? 1.0f : 0.0f;
            __builtin_nontemporal_store(r, ((v4f*)out) + i);
            c4 += stride_mod;
            if (c4 >= ncols4) c4 -= ncols4;
        }
    }
}

// Fallback for shapes whose row length is not a multiple of 4 (not hit by the
// reference 8192x4096 problem; keeps kernel_launch deterministic/general).
__global__ __launch_bounds__(256) void binarize_scalar_kernel(
    const float* __restrict__ in, const float* __restrict__ med,
    float* __restrict__ out, int n, int ncols)
{
    int i      = (int)(blockIdx.x * blockDim.x + threadIdx.x);
    int stride = (int)(gridDim.x * blockDim.x);
    for (; i < n; i += stride) {
        int c   = i % ncols;
        float m = med[c];
        out[i]  = (m > 0.0f && in[i] >= m) ? 1.0f : 0.0f;
    }
}

extern "C" void kernel_launch(void* const* d_in, const int* in_sizes, int n_in,
                              void* d_out, int out_size, void* d_ws, size_t ws_size,
                              hipStream_t stream) {
    (void)n_in; (void)d_ws; (void)ws_size; (void)out_size;

    const float* in  = (const float*)d_in[0];   // inputs  [8192, 4096] f32
    const float* med = (const float*)d_in[1];   // medians [4096]       f32
    float*       out = (float*)d_out;           // bool -> f32 (0.0 / 1.0)

    const int n     = in_sizes[0];              // 33,554,432
    const int ncols = in_sizes[1];              // 4,096

    const int threads = 256;

    if ((ncols % 4) != 0) {
        int blocks = (n + threads - 1) / threads;
        if (blocks > 4096) blocks = 4096;
        if (blocks < 1) blocks = 1;
        binarize_scalar_kernel<<<blocks, threads, 0, stream>>>(in, med, out, n, ncols);
        return;
    }

    const int total4 = n / 4;        // 8,388,608 float4s
    const int ncols4 = ncols / 4;    // 1,024 float4s per row

    int blocks = 2048;               // 524,288 threads -> 16 float4s each
    if ((long long)blocks * threads > total4) {
        blocks = (total4 + threads - 1) / threads;
        if (blocks < 1) blocks = 1;
    }
    const int stride     = blocks * threads;
    const int stride_mod = stride % ncols4;  // 0 for the reference shape

    binarize_vec_kernel<<<blocks, threads, 0, stream>>>(
        in, med, out, total4, ncols4, stride_mod);
}